// SSEMultiHeadAttention_17566416241403
// MI455X (gfx1250) — compile-verified
//
#include <hip/hip_runtime.h>

typedef __attribute__((ext_vector_type(2))) float v2f;
typedef __attribute__((ext_vector_type(8))) float v8f;

#define B_    2
#define S_    512
#define D_    1024
#define H_    16
#define HD_   64
#define P_    64
#define R_    4
#define TOPK_ 4
#define NROW_ (P_*R_)      // 256 state rows
#define LP_   65           // padded row stride (floats) to kill bank conflicts

// ---------------- LDS layout (floats) for the scan kernel ----------------
// st    : 256 rows x 65          (padded state)          65.0 KB
// psumT : 64 x 65 (psumT[e][p])  (transposed row-sums)   16.3 KB
// wq    : 64 x 64 (d-major)                              16   KB
// wkT   : 64 x 65 (wkT[e][d])    (transposed Wk)         16.3 KB
// wv    : 64 x 64 (d-major)                              16   KB
// bq,bv, xt,q,kq,sc,ctx : 7*64 ; lg,at : 2*16 ; sidx : 4
#define OFF_ST    0
#define OFF_PSUMT (OFF_ST    + NROW_*LP_)     // 16640
#define OFF_WQ    (OFF_PSUMT + HD_*LP_)       // 20800
#define OFF_WKT   (OFF_WQ    + HD_*HD_)       // 24896
#define OFF_WV    (OFF_WKT   + HD_*LP_)       // 29056
#define OFF_BQ    (OFF_WV    + HD_*HD_)       // 33152
#define OFF_BV    (OFF_BQ    + HD_)
#define OFF_XT    (OFF_BV    + HD_)
#define OFF_Q     (OFF_XT    + HD_)
#define OFF_KQ    (OFF_Q     + HD_)
#define OFF_SC    (OFF_KQ    + HD_)
#define OFF_CTX   (OFF_SC    + HD_)
#define OFF_LG    (OFF_CTX   + HD_)
#define OFF_AT    (OFF_LG    + TOPK_*R_)
#define OFF_IDX   (OFF_AT    + TOPK_*R_)
#define SMEM_FLOATS (OFF_IDX + TOPK_)

// One workgroup (64 threads = 2 wave32) per (b,h) stream; the 512-step
// recurrence runs entirely in-kernel with all state resident in LDS.
// Per-step K/V projections are algebraically folded:
//   logits[n] = G[n] . (Wk @ q)        (bk shift is softmax-invariant)
//   out       = (attn^T G) @ Wv + bv   (sum(attn) == 1)
__global__ __launch_bounds__(64) void sse_scan_kernel(
    const float* __restrict__ x,
    const float* __restrict__ Wq, const float* __restrict__ bq,
    const float* __restrict__ Wk,
    const float* __restrict__ Wv, const float* __restrict__ bv,
    const float* __restrict__ state0,
    float* __restrict__ y)                 // [B,S,D] pre-projection output
{
    extern __shared__ float sm[];
    float* st    = sm + OFF_ST;
    float* psumT = sm + OFF_PSUMT;
    float* wq    = sm + OFF_WQ;
    float* wkT   = sm + OFF_WKT;
    float* wv    = sm + OFF_WV;
    float* bqs   = sm + OFF_BQ;
    float* bvs   = sm + OFF_BV;
    float* xt    = sm + OFF_XT;
    float* q     = sm + OFF_Q;
    float* kq    = sm + OFF_KQ;
    float* sc    = sm + OFF_SC;
    float* ctx   = sm + OFF_CTX;
    float* lg    = sm + OFF_LG;
    float* at    = sm + OFF_AT;
    int*   sidx  = (int*)(sm + OFF_IDX);

    const int tid = threadIdx.x;            // 0..63, owns element/column tid
    const int bh  = blockIdx.x;             // 0..31
    const int b   = bh >> 4;
    const int h   = bh & 15;

    // ---- stage per-head weights (wkT transposed, stride-65 padded) ----
    for (int i = tid; i < HD_*HD_; i += 64) {
        const int d = i >> 6, e = i & 63;
        wq[i]           = Wq[h*HD_*HD_ + i];
        wv[i]           = Wv[h*HD_*HD_ + i];
        wkT[e*LP_ + d]  = Wk[h*HD_*HD_ + i];
    }
    bqs[tid] = bq[h*HD_ + tid];
    bvs[tid] = bv[h*HD_ + tid];

    // ---- stage initial state into padded rows ----
    const float* s0 = state0 + (size_t)bh * (NROW_*HD_);
    for (int i = tid; i < NROW_*HD_; i += 64) {
        const int row = i >> 6, d = i & 63;
        st[row*LP_ + d] = s0[i];
    }
    __syncthreads();

    // ---- transposed partition row-sums: psumT[e][p] = sum_r st[p,r,e] ----
    {
        const int e = tid;
        for (int p = 0; p < P_; ++p) {
            float s = 0.f;
#pragma unroll
            for (int r = 0; r < R_; ++r) s += st[(p*R_ + r)*LP_ + e];
            psumT[e*LP_ + p] = s;
        }
    }
    __syncthreads();

    for (int t = 0; t < S_; ++t) {
        // 1) token slice
        xt[tid] = x[((size_t)(b*S_ + t))*D_ + h*HD_ + tid];
        __syncthreads();

        // 2) q = xt @ Wq + bq         (lane e; wq reads are bank-clean)
        {
            float acc = bqs[tid];
            for (int d = 0; d < HD_; ++d) acc += xt[d] * wq[d*HD_ + tid];
            q[tid] = acc;
        }
        __syncthreads();

        // 3) scores[p] = q . mean(state[p])  AND  kq = Wk @ q   (fused pass)
        {
            float s = 0.f, ka = 0.f;
            for (int e = 0; e < HD_; ++e) {
                const float qe = q[e];                 // LDS broadcast
                s  += qe * psumT[e*LP_ + tid];         // lane p, bank-clean
                ka += qe * wkT  [e*LP_ + tid];         // lane d, bank-clean
            }
            sc[tid] = s * 0.25f;                       // mean = psum/R
            kq[tid] = ka;
        }
        __syncthreads();

        // 4) top-4 partitions (stable argmax == jax.lax.top_k tie-break)
        if (tid == 0) {
#pragma unroll
            for (int j = 0; j < TOPK_; ++j) {
                float best = -3.4e38f; int bi = 0;
                for (int p = 0; p < P_; ++p)
                    if (sc[p] > best) { best = sc[p]; bi = p; }
                sidx[j] = bi;
                sc[bi] = -3.4e38f;
            }
        }
        __syncthreads();

        // 5) logits[n] = scale * G[n].kq   (row padding -> distinct banks)
        if (tid < TOPK_*R_) {
            const int base = (sidx[tid >> 2]*R_ + (tid & 3)) * LP_;
            float acc = 0.f;
            for (int d = 0; d < HD_; ++d) acc += st[base + d] * kq[d];
            lg[tid] = acc * 0.125f;                    // 1/sqrt(64)
        }
        __syncthreads();

        // 6) softmax over 16
        if (tid == 0) {
            float m = lg[0];
            for (int n = 1; n < TOPK_*R_; ++n) m = fmaxf(m, lg[n]);
            float ssum = 0.f;
            for (int n = 0; n < TOPK_*R_; ++n) { float e = __expf(lg[n]-m); at[n]=e; ssum+=e; }
            const float inv = 1.f / ssum;
            for (int n = 0; n < TOPK_*R_; ++n) at[n] *= inv;
        }
        __syncthreads();

        // 7) ctx[d] = sum_n attn[n] * G[n][d]
        {
            float acc = 0.f;
#pragma unroll
            for (int n = 0; n < TOPK_*R_; ++n) {
                const int base = (sidx[n >> 2]*R_ + (n & 3)) * LP_;
                acc += at[n] * st[base + tid];
            }
            ctx[tid] = acc;
        }
        __syncthreads();

        // 8) out = ctx @ Wv + bv ; write y ; FIFO state + psum update
        {
            float o = bvs[tid];
            for (int d = 0; d < HD_; ++d) o += ctx[d] * wv[d*HD_ + tid];
            y[((size_t)(b*S_ + t))*D_ + h*HD_ + tid] = o;

            const int p  = sidx[0];
            const int pb = p*R_*LP_ + tid;
            const float old0 = st[pb];
            st[pb          ] = st[pb +   LP_];
            st[pb +   LP_  ] = st[pb + 2*LP_];
            st[pb + 2*LP_  ] = st[pb + 3*LP_];
            st[pb + 3*LP_  ] = xt[tid];
            psumT[tid*LP_ + p] += xt[tid] - old0;
        }
        __syncthreads();
    }
}

// Output projection: out[m,n] = sum_d y[m,d]*Wo[n,d] + bo[n]
// M=N=K=1024 fp32 via V_WMMA_F32_16X16X4_F32; 16x16 tile per wave32,
// 8 waves/block -> 32x64 per block; operands are L2-resident (8 MB << 192 MB).
__global__ __launch_bounds__(256) void out_proj_kernel(
    const float* __restrict__ y, const float* __restrict__ Wo,
    const float* __restrict__ bo, float* __restrict__ out)
{
    const int tid  = threadIdx.x;
    const int lane = tid & 31;
    const int wave = tid >> 5;
    const int lm   = lane & 15;
    const int lh   = lane >> 4;
    const int mTile = blockIdx.y*32 + (wave >> 2)*16;
    const int nTile = blockIdx.x*64 + (wave & 3)*16;

    const float* arow = y  + (size_t)(mTile + lm)*D_;   // A: row of y
    const float* brow = Wo + (size_t)(nTile + lm)*D_;   // B = Wo.T -> row of Wo

    v8f c = {};
#pragma unroll 8
    for (int k = 0; k < D_; k += 4) {
        v2f a  = *(const v2f*)&arow[k + 2*lh];
        v2f bf = *(const v2f*)&brow[k + 2*lh];
        c = __builtin_amdgcn_wmma_f32_16x16x4_f32(
                false, a, false, bf, (short)0, c, false, false);
    }
    const int col  = nTile + lm;
    const float bb = bo[col];
#pragma unroll
    for (int i = 0; i < 8; ++i) {
        const int row = mTile + i + 8*lh;
        out[(size_t)row*D_ + col] = c[i] + bb;
    }
}

extern "C" void kernel_launch(void* const* d_in, const int* in_sizes, int n_in,
                              void* d_out, int out_size, void* d_ws, size_t ws_size,
                              hipStream_t stream) {
    const float* x      = (const float*)d_in[0];
    const float* Wq     = (const float*)d_in[1];
    const float* bq     = (const float*)d_in[2];
    const float* Wk     = (const float*)d_in[3];
    const float* Wv     = (const float*)d_in[5];
    const float* bv     = (const float*)d_in[6];
    const float* Wo     = (const float*)d_in[7];
    const float* bo     = (const float*)d_in[8];
    const float* state0 = (const float*)d_in[9];
    float* y   = (float*)d_ws;                 // [B,S,D] scratch, 4 MB
    float* out = (float*)d_out;

    const size_t smemBytes = (size_t)SMEM_FLOATS * sizeof(float);  // ~134 KB < 320 KB WGP LDS
    (void)hipFuncSetAttribute((const void*)sse_scan_kernel,
                              hipFuncAttributeMaxDynamicSharedMemorySize,
                              (int)smemBytes);

    sse_scan_kernel<<<B_*H_, 64, smemBytes, stream>>>(
        x, Wq, bq, Wk, Wv, bv, state0, y);

    out_proj_kernel<<<dim3(D_/64, (B_*S_)/32), 256, 0, stream>>>(y, Wo, bo, out);
}